// Inverter_76141180224114
// MI455X (gfx1250) — compile-verified
//
#include <hip/hip_runtime.h>

#define T_LEN 16384
#define B_SZ  256
#define STRIDE (B_SZ * 2)   // floats per timestep row in [T][B][2] buffers

typedef __attribute__((ext_vector_type(2))) float v2f;
typedef __attribute__((ext_vector_type(8))) float v8f;

// ---------------- fast device transcendentals (gfx1250 TRANS unit) ----------

__device__ __forceinline__ float fast_rcp(float x) {
#if __has_builtin(__builtin_amdgcn_rcpf)
    return __builtin_amdgcn_rcpf(x);
#else
    return 1.0f / x;
#endif
}

__device__ __forceinline__ float fast_exp2(float x) {
#if __has_builtin(__builtin_amdgcn_exp2f)
    return __builtin_amdgcn_exp2f(x);
#else
    return exp2f(x);
#endif
}

__device__ __forceinline__ float act_tanh(float x) {
#if __has_builtin(__builtin_amdgcn_tanhf)
    return __builtin_amdgcn_tanhf(x);          // v_tanh_f32 (gfx1250)
#else
    // tanh(x) = 1 - 2/(exp2(2x*log2e)+1)
    float e = fast_exp2(x * 2.8853900817779268f);
    return 1.0f - 2.0f * fast_rcp(e + 1.0f);
#endif
}

__device__ __forceinline__ float act_sigm(float x) {
#if __has_builtin(__builtin_amdgcn_tanhf)
    return fmaf(0.5f, __builtin_amdgcn_tanhf(0.5f * x), 0.5f);
#else
    return fast_rcp(1.0f + fast_exp2(x * -1.4426950408889634f));
#endif
}

// ---------------- GRU layer 0: input = x [B][T], scalar input -------------
// One thread per batch element; runs fwd and bwd chains together (ILP=2).
__global__ void gru_l0(const float* __restrict__ x,
                       const float* __restrict__ wih,  // [2][3]
                       const float* __restrict__ whh,  // [2][3]
                       const float* __restrict__ bih,  // [2][3]
                       const float* __restrict__ bhh,  // [2][3]
                       float* __restrict__ out)        // [T][B][2]
{
    const int b = blockIdx.x * 32 + (threadIdx.x & 31);

    // forward (dir 0) weights
    const float f_wir = wih[0], f_wiz = wih[1], f_win = wih[2];
    const float f_whr = whh[0], f_whz = whh[1], f_whn = whh[2];
    const float f_br  = bih[0] + bhh[0];
    const float f_bz  = bih[1] + bhh[1];
    const float f_bin = bih[2], f_bhn = bhh[2];
    // backward (dir 1) weights
    const float g_wir = wih[3], g_wiz = wih[4], g_win = wih[5];
    const float g_whr = whh[3], g_whz = whh[4], g_whn = whh[5];
    const float g_br  = bih[3] + bhh[3];
    const float g_bz  = bih[4] + bhh[4];
    const float g_bin = bih[5], g_bhn = bhh[5];

    const float* xrow = x + (size_t)b * T_LEN;

    float hf = 0.0f, hb = 0.0f;
    float xf = xrow[0];
    float xb = xrow[T_LEN - 1];

    for (int t = 0; t < T_LEN; ++t) {
        const int tb = T_LEN - 1 - t;
        // software pipeline: next step's inputs
        const int tn = (t + 1 < T_LEN) ? (t + 1) : t;
        float nxf = xrow[tn];
        float nxb = xrow[T_LEN - 1 - tn];

        // forward chain
        {
            float r = act_sigm(fmaf(hf, f_whr, fmaf(xf, f_wir, f_br)));
            float z = act_sigm(fmaf(hf, f_whz, fmaf(xf, f_wiz, f_bz)));
            float n = act_tanh(fmaf(r, fmaf(hf, f_whn, f_bhn), fmaf(xf, f_win, f_bin)));
            hf = (1.0f - z) * n + z * hf;
            out[t * STRIDE + b * 2 + 0] = hf;
        }
        // backward chain
        {
            float r = act_sigm(fmaf(hb, g_whr, fmaf(xb, g_wir, g_br)));
            float z = act_sigm(fmaf(hb, g_whz, fmaf(xb, g_wiz, g_bz)));
            float n = act_tanh(fmaf(r, fmaf(hb, g_whn, g_bhn), fmaf(xb, g_win, g_bin)));
            hb = (1.0f - z) * n + z * hb;
            out[tb * STRIDE + b * 2 + 1] = hb;
        }
        xf = nxf;
        xb = nxb;
    }
}

// ---------------- GRU layers 1-2: input = [T][B][2] -----------------------
__global__ void gru_l12(const float* __restrict__ in,   // [T][B][2]
                        const float* __restrict__ wih,  // [2][3][2]
                        const float* __restrict__ whh,  // [2][3]
                        const float* __restrict__ bih,  // [2][3]
                        const float* __restrict__ bhh,  // [2][3]
                        float* __restrict__ out)        // [T][B][2]
{
    const int b = blockIdx.x * 32 + (threadIdx.x & 31);

    // forward (dir 0)
    const float f_wir0 = wih[0], f_wir1 = wih[1];
    const float f_wiz0 = wih[2], f_wiz1 = wih[3];
    const float f_win0 = wih[4], f_win1 = wih[5];
    const float f_whr = whh[0], f_whz = whh[1], f_whn = whh[2];
    const float f_br  = bih[0] + bhh[0];
    const float f_bz  = bih[1] + bhh[1];
    const float f_bin = bih[2], f_bhn = bhh[2];
    // backward (dir 1)
    const float g_wir0 = wih[6],  g_wir1 = wih[7];
    const float g_wiz0 = wih[8],  g_wiz1 = wih[9];
    const float g_win0 = wih[10], g_win1 = wih[11];
    const float g_whr = whh[3], g_whz = whh[4], g_whn = whh[5];
    const float g_br  = bih[3] + bhh[3];
    const float g_bz  = bih[4] + bhh[4];
    const float g_bin = bih[5], g_bhn = bhh[5];

    const float2* in2 = reinterpret_cast<const float2*>(in);

    float hf = 0.0f, hb = 0.0f;
    float2 xf = in2[0 * B_SZ + b];
    float2 xb = in2[(T_LEN - 1) * B_SZ + b];

    for (int t = 0; t < T_LEN; ++t) {
        const int tb = T_LEN - 1 - t;
        const int tn = (t + 1 < T_LEN) ? (t + 1) : t;
        float2 nxf = in2[tn * B_SZ + b];
        float2 nxb = in2[(T_LEN - 1 - tn) * B_SZ + b];

        // forward chain
        {
            float ir = fmaf(xf.y, f_wir1, fmaf(xf.x, f_wir0, f_br));
            float iz = fmaf(xf.y, f_wiz1, fmaf(xf.x, f_wiz0, f_bz));
            float in_ = fmaf(xf.y, f_win1, fmaf(xf.x, f_win0, f_bin));
            float r = act_sigm(fmaf(hf, f_whr, ir));
            float z = act_sigm(fmaf(hf, f_whz, iz));
            float n = act_tanh(fmaf(r, fmaf(hf, f_whn, f_bhn), in_));
            hf = (1.0f - z) * n + z * hf;
            out[t * STRIDE + b * 2 + 0] = hf;
        }
        // backward chain
        {
            float ir = fmaf(xb.y, g_wir1, fmaf(xb.x, g_wir0, g_br));
            float iz = fmaf(xb.y, g_wiz1, fmaf(xb.x, g_wiz0, g_bz));
            float in_ = fmaf(xb.y, g_win1, fmaf(xb.x, g_win0, g_bin));
            float r = act_sigm(fmaf(hb, g_whr, ir));
            float z = act_sigm(fmaf(hb, g_whz, iz));
            float n = act_tanh(fmaf(r, fmaf(hb, g_whn, g_bhn), in_));
            hb = (1.0f - z) * n + z * hb;
            out[tb * STRIDE + b * 2 + 1] = hb;
        }
        xf = nxf;
        xb = nxb;
    }
}

// ---------------- Output projection via V_WMMA_F32_16X16X4_F32 ------------
// y[i] = w0*in[i][0] + w1*in[i][1] + b  for i = t*B + b, written to y_tb[t*B+b].
// A (16x4, M across lanes): lanes 0-15 hold K0/K1 = {w0,w1}; lanes 16-31 K2/K3 = {b,0}.
// B (4x16, N across lanes): lanes 0-15 hold K0/K1 = data row; lanes 16-31 K2/K3 = {1,0}.
// D[m,n] = y[n] for all m; D VGPR0 = (M=0, N=lane) / (M=8, N=lane-16) -> direct store.
__global__ void proj_wmma(const float* __restrict__ in,     // [T][B][2]
                          const float* __restrict__ w_out,  // [2]
                          const float* __restrict__ b_out,  // [1]
                          float* __restrict__ y_tb)         // [T*B]
{
    const int lane   = threadIdx.x & 31;
    const int wid    = (blockIdx.x * blockDim.x + threadIdx.x) >> 5;
    const int nwaves = (gridDim.x * blockDim.x) >> 5;

    const float w0 = w_out[0], w1 = w_out[1], bo = b_out[0];

    v2f am;
    if (lane < 16) { am.x = w0; am.y = w1; }
    else           { am.x = bo; am.y = 0.0f; }

    const float2* in2 = reinterpret_cast<const float2*>(in);
    const int ntiles = (T_LEN * B_SZ) / 16;

    for (int tile = wid; tile < ntiles; tile += nwaves) {
        const int i0 = tile * 16;
        v2f bm;
        if (lane < 16) {
            float2 v = in2[i0 + lane];
            bm.x = v.x; bm.y = v.y;
        } else {
            bm.x = 1.0f; bm.y = 0.0f;
        }
        v8f c = {0.0f, 0.0f, 0.0f, 0.0f, 0.0f, 0.0f, 0.0f, 0.0f};
#if __has_builtin(__builtin_amdgcn_wmma_f32_16x16x4_f32)
        v8f d = __builtin_amdgcn_wmma_f32_16x16x4_f32(
            /*neg_a=*/false, am, /*neg_b=*/false, bm,
            /*c_mod=*/(short)0, c, /*reuse_a=*/false, /*reuse_b=*/false);
        // every lane holds y[i0 + (lane&15)] in d[0] (lanes 16-31 duplicate M=8 row)
        y_tb[i0 + (lane & 15)] = d[0];
#else
        // fallback (should not trigger on gfx1250 toolchains)
        float y = fmaf(bm.x, am.x, fmaf(bm.y, am.y, 0.0f));
        (void)c;
        if (lane < 16) y_tb[i0 + lane] = fmaf(w1, bm.y, fmaf(w0, bm.x, bo));
        (void)y;
#endif
    }
}

// ---------------- [T][B] -> [B][T] tiled transpose ------------------------
__global__ void transpose_k(const float* __restrict__ ytb, float* __restrict__ out)
{
    __shared__ float tile[32][33];
    const int tx = threadIdx.x, ty = threadIdx.y;
    const int t0 = blockIdx.x * 32, b0 = blockIdx.y * 32;
#pragma unroll
    for (int j = 0; j < 32; j += 8)
        tile[ty + j][tx] = ytb[(size_t)(t0 + ty + j) * B_SZ + (b0 + tx)];
    __syncthreads();
#pragma unroll
    for (int j = 0; j < 32; j += 8)
        out[(size_t)(b0 + ty + j) * T_LEN + (t0 + tx)] = tile[tx][ty + j];
}

// ---------------- host launcher -------------------------------------------
extern "C" void kernel_launch(void* const* d_in, const int* in_sizes, int n_in,
                              void* d_out, int out_size, void* d_ws, size_t ws_size,
                              hipStream_t stream)
{
    (void)in_sizes; (void)n_in; (void)out_size; (void)ws_size;

    const float* x      = (const float*)d_in[0];
    const float* w_ih0  = (const float*)d_in[1];
    const float* w_hh0  = (const float*)d_in[2];
    const float* b_ih0  = (const float*)d_in[3];
    const float* b_hh0  = (const float*)d_in[4];
    const float* w_ih12 = (const float*)d_in[5];
    const float* w_hh12 = (const float*)d_in[6];
    const float* b_ih12 = (const float*)d_in[7];
    const float* b_hh12 = (const float*)d_in[8];
    const float* w_out  = (const float*)d_in[9];
    const float* b_out  = (const float*)d_in[10];
    float* out = (float*)d_out;

    // scratch: two [T][B][2] ping-pong buffers (64 MiB total)
    float* bufA = (float*)d_ws;
    float* bufB = bufA + (size_t)T_LEN * B_SZ * 2;
    float* y_tb = bufB;   // reuse bufB after layer 2 has consumed it

    // 8 blocks x 32 threads: one wave32 per 32 batch elements, 2 chains/thread
    gru_l0 <<<B_SZ / 32, 32, 0, stream>>>(x, w_ih0, w_hh0, b_ih0, b_hh0, bufA);
    gru_l12<<<B_SZ / 32, 32, 0, stream>>>(bufA, w_ih12,      w_hh12,     b_ih12,     b_hh12,     bufB);
    gru_l12<<<B_SZ / 32, 32, 0, stream>>>(bufB, w_ih12 + 12, w_hh12 + 6, b_ih12 + 6, b_hh12 + 6, bufA);

    proj_wmma<<<512, 128, 0, stream>>>(bufA, w_out, b_out, y_tb);

    dim3 tgrid(T_LEN / 32, B_SZ / 32);
    dim3 tblk(32, 8);
    transpose_k<<<tgrid, tblk, 0, stream>>>(y_tb, out);
}